// DimensionAwareModulator_57561151701705
// MI455X (gfx1250) — compile-verified
//
#include <hip/hip_runtime.h>

#define NROWS 8192   // B*S = 16*512
#define DDIM  384
#define HDIM  64

typedef __attribute__((ext_vector_type(2))) float v2f;
typedef __attribute__((ext_vector_type(8))) float v8f;

__device__ __forceinline__ float relu1(float v) {
    // Single v_max_num_f32 (avoid clang's canonicalize max(x,x) + max(0,x)).
    float r;
    asm("v_max_num_f32 %0, %1, 0" : "=v"(r) : "v"(v));
    return r;
}

__device__ __forceinline__ float fast_tanh(float v) {
#if __has_builtin(__builtin_amdgcn_tanhf)
    return __builtin_amdgcn_tanhf(v);   // gfx1250 v_tanh_f32
#else
    return tanhf(v);
#endif
}

// Kernel 1: coeff[row,d] = tanh( sum_h relu(x*w1+b1)*w2 + b2 )
// Block: 32 d's x 256 rows, 256 threads (8 waves); wave handles 4 d's.
// Per (d, 16-row tile): 4x v_wmma_f32_16x16x4_f32 compute T[h,m]=w1[h]*x[m]+b1[h]
// (K0 = w1 (x) x, K1 = b1 (x) ones, K2/K3 = 0, C = 0); relu+dot(w2) on VALU.
__global__ __launch_bounds__(256) void dam_coeff_kernel(
    const float* __restrict__ xg,
    const float* __restrict__ w1g,
    const float* __restrict__ b1g,
    const float* __restrict__ w2g,
    const float* __restrict__ b2g,
    float* __restrict__ outc)
{
    __shared__ float lw1[32 * HDIM];
    __shared__ float lb1[32 * HDIM];
    __shared__ float lw2[32 * HDIM];

    const int tid      = threadIdx.x;
    const int d_base   = blockIdx.x * 32;
    const int row_base = blockIdx.y * 256;

    // Stage this block's weights (32 d's) into LDS, coalesced.
    {
        const int gbase = d_base * HDIM;
        #pragma unroll
        for (int i = 0; i < 8; ++i) {
            const int idx = tid + i * 256;
            lw1[idx] = w1g[gbase + idx];
            lb1[idx] = b1g[gbase + idx];
            lw2[idx] = w2g[gbase + idx];
        }
    }
    __syncthreads();

    const int wv   = tid >> 5;
    const int lane = tid & 31;
    const int hi   = lane >> 4;   // 0: lanes 0-15 (K0/K1 side), 1: lanes 16-31
    const int lm   = lane & 15;

    // Per-lane base pointers; all loop offsets become compile-time IOFFSETs.
    const float* xpw = xg   + (size_t)(row_base + lm) * DDIM + d_base + wv * 4;
    float*       opw = outc + (size_t)(row_base + lm) * DDIM + d_base + wv * 4;

    const float bones = hi ? 0.0f : 1.0f;   // B K1 row (ones), K3 = 0

    #pragma unroll
    for (int gi = 0; gi < 4; ++gi) {
        const int dl = wv * 4 + gi;          // wave-uniform
        const int d  = d_base + dl;

        // A-matrix per 16-wide h tile: lanes0-15 hold M=h rows; VGPR0=K0=w1,
        // VGPR1=K1=b1; lanes16-31 (K2/K3) zero.
        float a0[4], a1[4];
        #pragma unroll
        for (int t = 0; t < 4; ++t) {
            const float w  = lw1[dl * HDIM + t * 16 + lm];
            const float bb = lb1[dl * HDIM + t * 16 + lm];
            a0[t] = hi ? 0.0f : w;
            a1[t] = hi ? 0.0f : bb;
        }
        // Stage-2 weight per (tile, D-reg j, lane-half): w2[d, 16t + j + 8*hi]
        float wsel[4][8];
        #pragma unroll
        for (int t = 0; t < 4; ++t)
            #pragma unroll
            for (int j = 0; j < 8; ++j)
                wsel[t][j] = lw2[dl * HDIM + t * 16 + j + 8 * hi];

        const float b2v = b2g[d];            // wave-uniform -> s_load

        // Issue all 16 row-tile gathers up front (constant IOFFSETs).
        float xr[16];
        #pragma unroll
        for (int rt = 0; rt < 16; ++rt)
            xr[rt] = xpw[gi + rt * 16 * DDIM];

        #pragma unroll
        for (int rt = 0; rt < 16; ++rt) {
            // B-matrix 4x16: VGPR0 = [K0 | K2], VGPR1 = [K1 | K3].
            // K0 = x[m], K1 = ones, K2 = K3 = 0.
            v2f bmat;
            bmat.x = hi ? 0.0f : xr[rt];
            bmat.y = bones;

            // 4 independent accumulator chains (one per h tile).
            float acc[4];
            #pragma unroll
            for (int t = 0; t < 4; ++t) {
                v2f amat;
                amat.x = a0[t];
                amat.y = a1[t];
                v8f cz = {0.f, 0.f, 0.f, 0.f, 0.f, 0.f, 0.f, 0.f};
                // D[h=j+8*hi, m=lm] = w1[d,16t+h]*x[m] + b1[d,16t+h]
                v8f dmat = __builtin_amdgcn_wmma_f32_16x16x4_f32(
                    false, amat, false, bmat, (short)0, cz, false, false);
                float a = relu1(dmat[0]) * wsel[t][0];
                #pragma unroll
                for (int j = 1; j < 8; ++j)
                    a = fmaf(relu1(dmat[j]), wsel[t][j], a);
                acc[t] = a;
            }
            const float accs = (acc[0] + acc[1]) + (acc[2] + acc[3]);
            // lane-halves own disjoint halves of h -> one cross-half add
            const float other = __shfl_xor(accs, 16, 32);
            const float coeff = fast_tanh(accs + other + b2v);
            if (!hi) opw[gi + rt * 16 * DDIM] = coeff;
        }
    }
}

// Kernel 2: per row, modulated = coeff*base; scale by (std(base)+eps)/(std(mod)+eps)
// (unbiased, ddof=1). Reads coeff from d_out, rewrites d_out in place.
__global__ __launch_bounds__(128) void dam_scale_kernel(
    const float* __restrict__ base, float* __restrict__ out)
{
    const int row = blockIdx.x;
    const int tid = threadIdx.x;
    const float* bp = base + (size_t)row * DDIM;
    float* op = out + (size_t)row * DDIM;

    float m[3];
    float sb = 0.f, sb2 = 0.f, sm = 0.f, sm2 = 0.f;
    #pragma unroll
    for (int k = 0; k < 3; ++k) {
        const int i = tid + k * 128;
        const float c  = op[i];
        const float b  = bp[i];
        const float mm = c * b;
        m[k] = mm;
        sb += b;  sb2 += b * b;
        sm += mm; sm2 += mm * mm;
    }
    #pragma unroll
    for (int off = 16; off > 0; off >>= 1) {
        sb  += __shfl_xor(sb,  off, 32);
        sb2 += __shfl_xor(sb2, off, 32);
        sm  += __shfl_xor(sm,  off, 32);
        sm2 += __shfl_xor(sm2, off, 32);
    }
    __shared__ float red[4][4];
    const int wvid = tid >> 5;
    if ((tid & 31) == 0) {
        red[wvid][0] = sb;  red[wvid][1] = sb2;
        red[wvid][2] = sm;  red[wvid][3] = sm2;
    }
    __syncthreads();
    sb  = red[0][0] + red[1][0] + red[2][0] + red[3][0];
    sb2 = red[0][1] + red[1][1] + red[2][1] + red[3][1];
    sm  = red[0][2] + red[1][2] + red[2][2] + red[3][2];
    sm2 = red[0][3] + red[1][3] + red[2][3] + red[3][3];

    const float invD = 1.0f / 384.0f, invDm1 = 1.0f / 383.0f;
    const float varb = (sb2 - sb * sb * invD) * invDm1;
    const float varm = (sm2 - sm * sm * invD) * invDm1;
    const float scale = (sqrtf(fmaxf(varb, 0.f)) + 1e-6f) /
                        (sqrtf(fmaxf(varm, 0.f)) + 1e-6f);
    #pragma unroll
    for (int k = 0; k < 3; ++k)
        op[tid + k * 128] = m[k] * scale;
}

extern "C" void kernel_launch(void* const* d_in, const int* in_sizes, int n_in,
                              void* d_out, int out_size, void* d_ws, size_t ws_size,
                              hipStream_t stream) {
    const float* base = (const float*)d_in[0];
    const float* x    = (const float*)d_in[1];
    const float* w1   = (const float*)d_in[2];
    const float* b1   = (const float*)d_in[3];
    const float* w2   = (const float*)d_in[4];
    const float* b2   = (const float*)d_in[5];
    float* out = (float*)d_out;
    (void)in_sizes; (void)n_in; (void)out_size; (void)d_ws; (void)ws_size;

    dim3 g1(DDIM / 32, NROWS / 256);   // 12 x 32 blocks
    dam_coeff_kernel<<<g1, 256, 0, stream>>>(x, w1, b1, w2, b2, out);
    dam_scale_kernel<<<NROWS, 128, 0, stream>>>(base, out);
}